// TokenEncoder_90881507983845
// MI455X (gfx1250) — compile-verified
//
#include <hip/hip_runtime.h>

typedef __attribute__((ext_vector_type(2))) float v2f;
typedef __attribute__((ext_vector_type(8))) float v8f;

#define BSZ   4
#define LSEQ  1024
#define DIN   64
#define DMOD  512
#define NSIG  32
#define NROLE 3
#define NKEY  (NSIG * NROLE)        // 96
#define NTOK  (BSZ * LSEQ)          // 4096
#define MAXG  384                   // >= 256 + 95 worst-case groups

// -------- workspace layout (ints) --------
// order  : [0, 4096)
// gPkey  : [4096, 4096+384)
// gStart : [4480, 4864)
// gCount : [4864, 5248)
// numG   : [5248]

// ---------------------------------------------------------------------------
// Kernel 1: bin tokens by pkey = sid*3 + role; build groups of <=16 tokens.
// Single block; 4096 tokens is trivial.
// ---------------------------------------------------------------------------
__global__ __launch_bounds__(256) void te_binning(
    const int* __restrict__ sid, const int* __restrict__ role,
    int* __restrict__ order, int* __restrict__ gPkey,
    int* __restrict__ gStart, int* __restrict__ gCount,
    int* __restrict__ numGroups)
{
    __shared__ int cnt[NKEY];
    __shared__ int off[NKEY];
    __shared__ int cur[NKEY];
    const int tid = threadIdx.x;

    if (tid < NKEY) cnt[tid] = 0;
    __syncthreads();

    for (int t = tid; t < NTOK; t += 256) {
        int pk = sid[t] * NROLE + role[t];
        atomicAdd(&cnt[pk], 1);
    }
    __syncthreads();

    if (tid == 0) {
        int running = 0, ng = 0;
        for (int k = 0; k < NKEY; ++k) {
            off[k] = running;
            int c = cnt[k];
            for (int s = 0; s < c; s += 16) {
                gPkey[ng]  = k;
                gStart[ng] = running + s;
                gCount[ng] = (c - s < 16) ? (c - s) : 16;
                ++ng;
            }
            running += c;
        }
        *numGroups = ng;
    }
    __syncthreads();

    if (tid < NKEY) cur[tid] = off[tid];
    __syncthreads();

    for (int t = tid; t < NTOK; t += 256) {
        int pk = sid[t] * NROLE + role[t];
        int slot = atomicAdd(&cur[pk], 1);
        order[slot] = t;
    }
}

// ---------------------------------------------------------------------------
// Kernel 2: per group of <=16 same-pkey tokens, 16x64 @ 64x512 GEMM via
// v_wmma_f32_16x16x4_f32. 256 threads = 8 waves; wave w owns N in
// [64w, 64w+64) as 4 16-wide accumulator tiles. Epilogue: bias, padding
// mask, four embedding gathers, scatter rows to tokens[b, l+1, :].
// ---------------------------------------------------------------------------
__global__ __launch_bounds__(256) void te_gemm(
    const float* __restrict__ emb,  const int* __restrict__ pos,
    const int* __restrict__ sid,    const int* __restrict__ mod,
    const int* __restrict__ role,   const int* __restrict__ pmask,
    const float* __restrict__ W,    const float* __restrict__ pb,
    const float* __restrict__ posE, const float* __restrict__ idE,
    const float* __restrict__ modE, const float* __restrict__ roleE,
    const int* __restrict__ order,  const int* __restrict__ gPkey,
    const int* __restrict__ gStart, const int* __restrict__ gCount,
    const int* __restrict__ numGroups, float* __restrict__ out)
{
    __shared__ float ldsA[16 * 68];          // 16 rows x 64 K, stride 68 (bank-conflict pad)
    __shared__ int sTok[16], sMask[16], sPos[16], sSid[16], sMod[16], sRole[16];

    const int g = blockIdx.x;
    if (g >= *numGroups) return;             // uniform across block

    const int tid = threadIdx.x;
    const int pk    = gPkey[g];
    const int start = gStart[g];
    const int count = gCount[g];

    if (tid < 16) {
        int tokv = (tid < count) ? order[start + tid] : -1;
        sTok[tid] = tokv;
        if (tokv >= 0) {
            sMask[tid] = pmask[tokv];
            sPos[tid]  = pos[tokv];
            sSid[tid]  = sid[tokv];
            sMod[tid]  = mod[tokv];
            sRole[tid] = role[tokv];
        } else {
            sMask[tid] = 0; sPos[tid] = 0; sSid[tid] = 0; sMod[tid] = 0; sRole[tid] = 0;
        }
    }
    __syncthreads();

    // Stage A tile: 16x64 f32 (zero-pad missing rows).
    #pragma unroll
    for (int i = 0; i < 4; ++i) {
        int idx = tid + 256 * i;
        int m = idx >> 6, k = idx & 63;
        int tokv = sTok[m];
        ldsA[m * 68 + k] = (tokv >= 0) ? emb[(size_t)tokv * DIN + k] : 0.0f;
    }
    __syncthreads();

    const int wave = tid >> 5, lane = tid & 31;
    const int half = lane >> 4, li = lane & 15;
    const int n0 = wave * 64;
    const float* __restrict__ Wp = W + (size_t)pk * (DIN * DMOD);

    v8f c0 = {}, c1 = {}, c2 = {}, c3 = {};

    #pragma unroll
    for (int k0 = 0; k0 < DIN; k0 += 4) {
        const int kk = k0 + 2 * half;        // A/B fragment K per lane-half (ISA 7.12.2)
        v2f a;
        a.x = ldsA[li * 68 + kk];
        a.y = ldsA[li * 68 + kk + 1];

        const float* wr0 = Wp + (size_t)kk * DMOD + n0 + li;   // row K=kk
        const float* wr1 = wr0 + DMOD;                         // row K=kk+1
        v2f b0, b1, b2, b3;
        b0.x = wr0[0];   b0.y = wr1[0];
        b1.x = wr0[16];  b1.y = wr1[16];
        b2.x = wr0[32];  b2.y = wr1[32];
        b3.x = wr0[48];  b3.y = wr1[48];

        c0 = __builtin_amdgcn_wmma_f32_16x16x4_f32(false, a, false, b0, (short)0, c0, false, false);
        c1 = __builtin_amdgcn_wmma_f32_16x16x4_f32(false, a, false, b1, (short)0, c1, false, false);
        c2 = __builtin_amdgcn_wmma_f32_16x16x4_f32(false, a, false, b2, (short)0, c2, false, false);
        c3 = __builtin_amdgcn_wmma_f32_16x16x4_f32(false, a, false, b3, (short)0, c3, false, false);
    }

    v8f cs[4] = { c0, c1, c2, c3 };

    #pragma unroll
    for (int t = 0; t < 4; ++t) {
        const int n = n0 + t * 16 + li;
        const float bn = pb[pk * DMOD + n];
        #pragma unroll
        for (int r = 0; r < 8; ++r) {
            const int m = r + 8 * half;      // C layout: VGPR r -> row r (+8 for upper half)
            const int tokv = sTok[m];
            if (tokv < 0) continue;
            float v = sMask[m] ? (cs[t][r] + bn) : 0.0f;
            v += posE[(size_t)sPos[m] * DMOD + n];
            v += idE [(size_t)sSid[m] * DMOD + n];
            v += modE[(size_t)sMod[m] * DMOD + n];
            v += roleE[(size_t)sRole[m] * DMOD + n];
            const int b = tokv >> 10;        // tokv / LSEQ
            const int l = tokv & (LSEQ - 1);
            out[((size_t)b * (LSEQ + 1) + 1 + l) * DMOD + n] = v;
        }
    }
}

// ---------------------------------------------------------------------------
// Kernel 3: CLS rows + attn_keep (stored as 0/1 floats after the tokens).
// ---------------------------------------------------------------------------
__global__ __launch_bounds__(256) void te_cls_attn(
    const float* __restrict__ cls, const float* __restrict__ posE,
    const float* __restrict__ idE, const int* __restrict__ pmask,
    float* __restrict__ out)
{
    const int i = blockIdx.x * blockDim.x + threadIdx.x;
    if (i < BSZ * DMOD) {
        int b = i / DMOD, n = i % DMOD;
        out[(size_t)b * (LSEQ + 1) * DMOD + n] =
            cls[n] + posE[n] + idE[(size_t)NSIG * DMOD + n];
    }
    if (i < BSZ * (LSEQ + 1)) {
        int b = i / (LSEQ + 1), j = i % (LSEQ + 1);
        float v = (j == 0) ? 1.0f : (pmask[b * LSEQ + j - 1] ? 1.0f : 0.0f);
        out[(size_t)BSZ * (LSEQ + 1) * DMOD + (size_t)b * (LSEQ + 1) + j] = v;
    }
}

// ---------------------------------------------------------------------------
extern "C" void kernel_launch(void* const* d_in, const int* in_sizes, int n_in,
                              void* d_out, int out_size, void* d_ws, size_t ws_size,
                              hipStream_t stream) {
    const float* emb   = (const float*)d_in[0];
    const int*   pos   = (const int*)  d_in[1];
    const int*   sid   = (const int*)  d_in[2];
    const int*   mod   = (const int*)  d_in[3];
    const int*   role  = (const int*)  d_in[4];
    const int*   pmask = (const int*)  d_in[5];
    const float* projW = (const float*)d_in[6];
    const float* projB = (const float*)d_in[7];
    const float* cls   = (const float*)d_in[8];
    const float* posE  = (const float*)d_in[9];
    const float* idE   = (const float*)d_in[10];
    const float* modE  = (const float*)d_in[11];
    const float* roleE = (const float*)d_in[12];
    float* out = (float*)d_out;

    int* ws     = (int*)d_ws;
    int* order  = ws;
    int* gPkey  = ws + NTOK;
    int* gStart = gPkey + MAXG;
    int* gCount = gStart + MAXG;
    int* numG   = gCount + MAXG;

    te_binning<<<1, 256, 0, stream>>>(sid, role, order, gPkey, gStart, gCount, numG);

    te_gemm<<<MAXG, 256, 0, stream>>>(emb, pos, sid, mod, role, pmask,
                                      projW, projB, posE, idE, modE, roleE,
                                      order, gPkey, gStart, gCount, numG, out);

    te_cls_attn<<<(BSZ * (LSEQ + 1) + 255) / 256, 256, 0, stream>>>(cls, posE, idE, pmask, out);
}